// VoronoiPropagation_65266323030072
// MI455X (gfx1250) — compile-verified
//
#include <hip/hip_runtime.h>
#include <math.h>

#define BATCH 16
#define NCLUST 64
#define HH 256
#define WW 256
#define PGRID 64   // pooled grid is 64x64 (H/4 x W/4)
#define PSZ 4      // 4x4 mean pool

typedef __attribute__((ext_vector_type(2))) float v2f;
typedef __attribute__((ext_vector_type(8))) float v8f;

// ---------------- Kernel 1: Sobel -> 1 - 0.5*|grad| -> 4x4 mean pool ----------------
// One thread per pooled cell; reads its 6x6 halo patch, computes 16 Sobel taps.
__global__ void k_pool(const float* __restrict__ x, float* __restrict__ pooled) {
    int g = blockIdx.x * blockDim.x + threadIdx.x;          // 0 .. B*64*64-1
    if (g >= BATCH * PGRID * PGRID) return;
    int b    = g / (PGRID * PGRID);
    int cell = g % (PGRID * PGRID);
    int py = cell / PGRID, px = cell % PGRID;
    const float* xb = x + (size_t)b * HH * WW;
    int r0 = py * PSZ - 1, c0 = px * PSZ - 1;
    float p[6][6];
#pragma unroll
    for (int dr = 0; dr < 6; ++dr) {
        int rr = r0 + dr;
#pragma unroll
        for (int dc = 0; dc < 6; ++dc) {
            int cc = c0 + dc;
            bool ok = (rr >= 0) && (rr < HH) && (cc >= 0) && (cc < WW);
            p[dr][dc] = ok ? xb[rr * WW + cc] : 0.0f;       // zero padding
        }
    }
    float acc = 0.0f;
#pragma unroll
    for (int i = 0; i < PSZ; ++i) {
#pragma unroll
        for (int j = 0; j < PSZ; ++j) {
            float gx = -p[i][j]       + p[i][j+2]
                     - 2.f*p[i+1][j]  + 2.f*p[i+1][j+2]
                     - p[i+2][j]      + p[i+2][j+2];
            float gy = -p[i][j] - 2.f*p[i][j+1] - p[i][j+2]
                     +  p[i+2][j] + 2.f*p[i+2][j+1] + p[i+2][j+2];
            float gm = sqrtf(gx * gx + gy * gy);
            acc += 1.0f - 0.5f * gm;
        }
    }
    pooled[g] = acc * (1.0f / 16.0f);
}

// ---------------- Kernel 2: per-batch top-64 (sorted desc, tie -> smaller idx) -------
__global__ void k_topk(const float* __restrict__ pooled,
                       float* __restrict__ sy, float* __restrict__ sx) {
    __shared__ float vals[PGRID * PGRID];
    __shared__ float rv[256];
    __shared__ int   ri[256];
    int b = blockIdx.x, tid = threadIdx.x;
    const float* pb = pooled + b * PGRID * PGRID;
    for (int i = tid; i < PGRID * PGRID; i += 256) vals[i] = pb[i];
    __syncthreads();
    for (int sel = 0; sel < NCLUST; ++sel) {
        float bv = -__builtin_inff();
        int   bi = PGRID * PGRID;
        for (int i = tid; i < PGRID * PGRID; i += 256) {
            float v = vals[i];
            if (v > bv) { bv = v; bi = i; }                 // strict > keeps smallest i
        }
        rv[tid] = bv; ri[tid] = bi;
        __syncthreads();
        for (int s = 128; s > 0; s >>= 1) {
            if (tid < s) {
                float ov = rv[tid + s]; int oi = ri[tid + s];
                if (ov > rv[tid] || (ov == rv[tid] && oi < ri[tid])) {
                    rv[tid] = ov; ri[tid] = oi;
                }
            }
            __syncthreads();
        }
        if (tid == 0) {
            int idx = ri[0];
            sy[b * NCLUST + sel] = (float)(idx / PGRID) * (1.0f / (float)NCLUST);
            sx[b * NCLUST + sel] = (float)(idx % PGRID) * (1.0f / (float)NCLUST);
            vals[idx] = -__builtin_inff();
        }
        __syncthreads();
    }
}

// ---------------- Kernel 3: axis factors Fy/Fx + softmax denominator -----------------
// Fy[b,r,k] = exp(0.5*exp(-((r/H - sy)/std)^2)) ; denom = (sum Fy)*(sum Fx)
__global__ void k_axes(const float* __restrict__ sy, const float* __restrict__ sx,
                       const float* __restrict__ stdp,
                       float* __restrict__ Fy, float* __restrict__ Fx,
                       float* __restrict__ recip) {
    __shared__ float ssy[256], ssx[256];
    int bk = blockIdx.x, t = threadIdx.x;                   // bk = b*64 + k
    float invstd = 1.0f / stdp[0];
    float cy = sy[bk], cx = sx[bk];
    float pos = (float)t * (1.0f / (float)HH);              // H == W
    float dy = (pos - cy) * invstd;
    float dx = (pos - cx) * invstd;
    float fy = expf(0.5f * expf(-dy * dy));
    float fx = expf(0.5f * expf(-dx * dx));
    Fy[bk * HH + t] = fy;
    Fx[bk * WW + t] = fx;
    ssy[t] = fy; ssx[t] = fx;
    __syncthreads();
    for (int s = 128; s > 0; s >>= 1) {
        if (t < s) { ssy[t] += ssy[t + s]; ssx[t] += ssx[t + s]; }
        __syncthreads();
    }
    if (t == 0) recip[bk] = 1.0f / (ssy[0] * ssx[0]);
}

// ---------------- Kernel 4: rank-1 outer product via WMMA, 256MB NT store ------------
// One wave per 16x256 strip of one output channel; 16x v_wmma_f32_16x16x4_f32.
// Fragment loads are unconditional (lanes 16-31 read the lane&15 duplicate) and the
// upper half is zeroed with a branchless select -> v_cndmask, no exec-mask branches.
__global__ void __launch_bounds__(256) k_outer(const float* __restrict__ Fy,
                                               const float* __restrict__ Fx,
                                               const float* __restrict__ recip,
                                               float* __restrict__ out) {
    int lane  = threadIdx.x & 31;
    int half  = lane & 15;
    bool lo   = (lane < 16);
    int wave  = threadIdx.x >> 5;
    int strip = blockIdx.x * 8 + wave;   // 0 .. B*K*(H/16)-1 = 16383
    int bk = strip >> 4;                 // channel index b*64+k
    int rb = strip & 15;                 // 16-row block within channel
    float rc = recip[bk];

    // A-matrix 16x4 f32: VGPR0 lanes0-15 = K=0 column (Fy segment * recip), rest zero.
    float a0 = Fy[bk * HH + rb * 16 + half] * rc;   // unconditional, coalesced
    a0 = lo ? a0 : 0.0f;                            // v_cndmask, no branch
    v2f A = {a0, 0.0f};

    const float* fxp = Fx + bk * WW;

    // Pre-load all 16 B fragments unconditionally so they can issue as one clause.
    float bfrag[16];
#pragma unroll
    for (int ct = 0; ct < 16; ++ct) bfrag[ct] = fxp[ct * 16 + half];

    float* op = out + ((size_t)bk * HH + (size_t)rb * 16) * WW;
#pragma unroll
    for (int ct = 0; ct < 16; ++ct) {
        // B-matrix 4x16 f32: VGPR0 lanes0-15 = row K=0 (Fx segment), rest zero.
        float b0 = lo ? bfrag[ct] : 0.0f;
        v2f Bm = {b0, 0.0f};
        v8f C = {0.f, 0.f, 0.f, 0.f, 0.f, 0.f, 0.f, 0.f};
        v8f D = __builtin_amdgcn_wmma_f32_16x16x4_f32(
            /*neg_a=*/false, A, /*neg_b=*/false, Bm,
            /*c_mod=*/(short)0, C, /*reuse_a=*/false, /*reuse_b=*/false);
        // D layout: VGPR j -> (M=j, N=lane) lanes 0-15; (M=j+8, N=lane-16) lanes 16-31
        int rowoff = (lane >> 4) << 3;
        int col = ct * 16 + half;
#pragma unroll
        for (int j = 0; j < 8; ++j) {
            __builtin_nontemporal_store(D[j], &op[(size_t)(rowoff + j) * WW + col]);
        }
    }
}

extern "C" void kernel_launch(void* const* d_in, const int* in_sizes, int n_in,
                              void* d_out, int out_size, void* d_ws, size_t ws_size,
                              hipStream_t stream) {
    const float* x    = (const float*)d_in[0];   // [16,1,256,256]
    const float* stdp = (const float*)d_in[1];   // scalar
    float* out = (float*)d_out;                  // [16,64,256,256]
    float* ws  = (float*)d_ws;

    // workspace layout (floats): total ~2.27 MB
    float* pooled = ws;                                  // 16*64*64   = 65536
    float* sy     = pooled + BATCH * PGRID * PGRID;      // 1024
    float* sx     = sy + BATCH * NCLUST;                 // 1024
    float* recip  = sx + BATCH * NCLUST;                 // 1024
    float* Fy     = recip + BATCH * NCLUST;              // 16*64*256 = 262144
    float* Fx     = Fy + BATCH * NCLUST * HH;            // 262144

    k_pool <<<(BATCH * PGRID * PGRID) / 256, 256, 0, stream>>>(x, pooled);
    k_topk <<<BATCH, 256, 0, stream>>>(pooled, sy, sx);
    k_axes <<<BATCH * NCLUST, 256, 0, stream>>>(sy, sx, stdp, Fy, Fx, recip);
    k_outer<<<(BATCH * NCLUST * (HH / 16)) / 8, 256, 0, stream>>>(Fy, Fx, recip, out);
}